// TrainableCloud_81423989997805
// MI455X (gfx1250) — compile-verified
//
#include <hip/hip_runtime.h>
#include <math.h>

#define N_STEPS 256
#define TPR 8                       // threads per ray
#define STEPS_PER_THREAD (N_STEPS / TPR)   // 32
#define DENSITY_SCALE 10.0f
#define EPSF 1e-8f

// grid is 256 x 256 x 256, row stride 256 floats (1024 B), slice stride 65536 floats (262144 B)

__device__ __forceinline__ float safe_inv(float d) {
    // reference: 1 / where(|d|<EPS, sign(d)*EPS+EPS, d)
    float s   = (d > 0.0f) ? 1.0f : ((d < 0.0f) ? -1.0f : 0.0f);
    float den = (fabsf(d) < EPSF) ? fmaf(s, EPSF, EPSF) : d;
    return 1.0f / den;
}

__device__ __forceinline__ float clamp01(float v) {
    return fminf(fmaxf(v, 0.0f), 1.0f);   // -> v_med3_f32
}

__global__ __launch_bounds__(256) void transmittance_kernel(
    const float* __restrict__ rays,
    const float* __restrict__ grid,
    float* __restrict__ out,
    int n_rays)
{
    const int tid   = blockIdx.x * blockDim.x + threadIdx.x;
    const int ray   = tid >> 3;        // TPR = 8
    const int chunk = tid & 7;
    if (ray >= n_rays) return;

    // ray row = 6 floats, 24-byte stride (compiler merges into wide loads; HW tolerates align-4)
    const float2* rp = (const float2*)(rays + ray * 6);
    float2 p0 = rp[0];   // ox, oy
    float2 p1 = rp[1];   // oz, dx
    float2 p2 = rp[2];   // dy, dz
    float ox = p0.x, oy = p0.y, oz = p1.x;
    float dx = p1.y, dy = p2.x, dz = p2.y;

    // normalize: d / (||d|| + eps)
    float nrm = sqrtf(fmaf(dx, dx, fmaf(dy, dy, dz * dz)));
    float inl = 1.0f / (nrm + EPSF);
    dx *= inl; dy *= inl; dz *= inl;

    // slab intersection with [-1,1]^3
    float ix = safe_inv(dx), iy = safe_inv(dy), iz = safe_inv(dz);
    float tax = (-1.0f - ox) * ix, tbx = (1.0f - ox) * ix;
    float tay = (-1.0f - oy) * iy, tby = (1.0f - oy) * iy;
    float taz = (-1.0f - oz) * iz, tbz = (1.0f - oz) * iz;
    float tmin = fmaxf(fmaxf(fminf(tax, tbx), fminf(tay, tby)), fminf(taz, tbz));
    float tmax = fminf(fminf(fmaxf(tax, tbx), fmaxf(tay, tby)), fmaxf(taz, tbz));
    tmin = fmaxf(tmin, 0.0f);
    bool  valid = tmax > tmin;
    float seg   = valid ? (tmax - tmin) : 0.0f;
    float dt    = seg * (1.0f / (float)N_STEPS);

    // voxel coord of sample i: g = A + (i+0.5)*B  (analytic -> no drift, no loop dep)
    float Ax = fmaf(fmaf(tmin, dx, ox), 127.5f, 127.5f);
    float Ay = fmaf(fmaf(tmin, dy, oy), 127.5f, 127.5f);
    float Az = fmaf(fmaf(tmin, dz, oz), 127.5f, 127.5f);
    float Bx = dt * dx * 127.5f;
    float By = dt * dy * 127.5f;
    float Bz = dt * dz * 127.5f;

    const float fbase = (float)(chunk * STEPS_PER_THREAD) + 0.5f;
    const char* gbytes = (const char*)grid;
    float sum = 0.0f;

    for (int i = 0; i < STEPS_PER_THREAD; i += 4) {
        // gfx1250 global_prefetch_b8: warm both z-slice lines 8 steps ahead along the ray.
        // Second prefetch reuses the same address register (constant +262144 offset).
        {
            float fp  = fbase + (float)(i + 8);
            float gxp = fmaf(fp, Bx, Ax);
            float gyp = fmaf(fp, By, Ay);
            float gzp = fmaf(fp, Bz, Az);
            float xpf = fminf(fmaxf(floorf(gxp), 0.0f), 254.0f);
            float ypf = fminf(fmaxf(floorf(gyp), 0.0f), 254.0f);
            float zpf = fminf(fmaxf(floorf(gzp), 0.0f), 254.0f);
            unsigned bofp = ((unsigned)fmaf(zpf, 65536.0f, fmaf(ypf, 256.0f, xpf))) << 2;
            __builtin_prefetch(gbytes + bofp, 0, 3);            // z0 slice line
            __builtin_prefetch(gbytes + bofp + 262144, 0, 3);   // z1 slice line
        }
#pragma unroll
        for (int j = 0; j < 4; ++j) {
            float fi = fbase + (float)(i + j);
            float gx = fmaf(fi, Bx, Ax);
            float gy = fmaf(fi, By, Ay);
            float gz = fmaf(fi, Bz, Az);

            float x0f = fminf(fmaxf(floorf(gx), 0.0f), 254.0f);
            float y0f = fminf(fmaxf(floorf(gy), 0.0f), 254.0f);
            float z0f = fminf(fmaxf(floorf(gz), 0.0f), 254.0f);
            float fx = fminf(fmaxf(gx - x0f, 0.0f), 1.0f);
            float fy = fminf(fmaxf(gy - y0f, 0.0f), 1.0f);
            float fz = fminf(fmaxf(gz - z0f, 0.0f), 1.0f);

            // exact in f32: idx <= 16,711,422 < 2^24 ; keep offset 32-bit for saddr-form loads
            unsigned bof = ((unsigned)fmaf(z0f, 65536.0f, fmaf(y0f, 256.0f, x0f))) << 2;

            // 4 x global_load_b64 (x-adjacent corner pairs), one per-lane 32-bit offset,
            // constant byte offsets fold into the 24-bit immediate.
            float2 a00 = *(const float2*)(gbytes + bof);            // z0,y0 : c000,c001
            float2 a01 = *(const float2*)(gbytes + bof + 1024);     // z0,y1 : c010,c011
            float2 a10 = *(const float2*)(gbytes + bof + 262144);   // z1,y0 : c100,c101
            float2 a11 = *(const float2*)(gbytes + bof + 263168);   // z1,y1 : c110,c111

            float c000 = clamp01(a00.x), c001 = clamp01(a00.y);
            float c010 = clamp01(a01.x), c011 = clamp01(a01.y);
            float c100 = clamp01(a10.x), c101 = clamp01(a10.y);
            float c110 = clamp01(a11.x), c111 = clamp01(a11.y);

            float c00 = fmaf(fx, c001 - c000, c000);
            float c01 = fmaf(fx, c011 - c010, c010);
            float c10 = fmaf(fx, c101 - c100, c100);
            float c11 = fmaf(fx, c111 - c110, c110);
            float c0  = fmaf(fy, c01 - c00, c00);
            float c1  = fmaf(fy, c11 - c10, c10);
            sum += fmaf(fz, c1 - c0, c0);
        }
    }

    // reduce the 8 chunk-partials of this ray (lanes r*8 .. r*8+7 of the wave32)
    sum += __shfl_xor(sum, 1, 32);
    sum += __shfl_xor(sum, 2, 32);
    sum += __shfl_xor(sum, 4, 32);

    if (chunk == 0) {
        float tau = DENSITY_SCALE * sum * dt;
        float T   = valid ? __expf(-tau) : 1.0f;
        // result is write-once, never re-read: NT store keeps L2 for the grid
        __builtin_nontemporal_store(T, out + ray);
    }
}

extern "C" void kernel_launch(void* const* d_in, const int* in_sizes, int n_in,
                              void* d_out, int out_size, void* d_ws, size_t ws_size,
                              hipStream_t stream) {
    const float* rays = (const float*)d_in[0];
    const float* grid = (const float*)d_in[1];
    float* out = (float*)d_out;

    const int n_rays  = in_sizes[0] / 6;          // 65536
    const int threads = n_rays * TPR;             // 524288
    dim3 block(256);
    dim3 grid_dim((threads + 255) / 256);         // 2048 blocks
    transmittance_kernel<<<grid_dim, block, 0, stream>>>(rays, grid, out, n_rays);
}